// FourierFlow_44650480009357
// MI455X (gfx1250) — compile-verified
//
#include <hip/hip_runtime.h>

typedef __attribute__((ext_vector_type(2))) float v2f;
typedef __attribute__((ext_vector_type(4))) float v4f;
typedef __attribute__((ext_vector_type(8))) float v8f;

#define TPW 16                 // n-tiles per wave
#define NWAVES 8               // waves per block (256 threads, wave32)
#define TPB (TPW * NWAVES)     // n-tiles per block

// Wave-level 16x16x16 fp32 matmul: D = A x B, A/B row-major 16x16 in LDS.
// Built from 4 chained V_WMMA_F32_16X16X4_F32 (k-chunks of 4).
__device__ inline v8f wave_mm16(const float* Am, const float* Bm, int r, int h) {
  v8f c = {};
#pragma unroll
  for (int kk = 0; kk < 4; ++kk) {
    const int kb = 4 * kk + 2 * h;
    v2f a, b;
    a.x = Am[r * 16 + kb];
    a.y = Am[r * 16 + kb + 1];
    b.x = Bm[kb * 16 + r];
    b.y = Bm[(kb + 1) * 16 + r];
    c = __builtin_amdgcn_wmma_f32_16x16x4_f32(false, a, false, b, (short)0, c,
                                              false, false);
  }
  return c;
}

// Scatter a 16x16 f32 C/D fragment back to row-major LDS.
__device__ inline void frag_to_lds(float* Dm, v8f c, int r, int h) {
#pragma unroll
  for (int q = 0; q < 8; ++q) Dm[(q + 8 * h) * 16 + r] = c[q];
}

__global__ __launch_bounds__(256) void fourierflow_kernel(
    const float* __restrict__ x, const float* __restrict__ tt,
    const float* __restrict__ omega, const float* __restrict__ M,
    const float* __restrict__ M0, float* __restrict__ out, int n_points,
    int n_tiles) {
  __shared__ float sB[256];  // scaled generator A/2^8
  __shared__ float sT[256];  // Taylor / squaring ping
  __shared__ float sU[256];  // squaring pong
  const int tid = threadIdx.x;
  const int lane = tid & 31;
  const int wave = tid >> 5;
  const int r = lane & 15;   // A-row / B-col / D-col index for this lane
  const int h = lane >> 4;   // half-select
  const int ti = blockIdx.y; // time index

  // ---- wave 0: E = expm((M*omega + M0) * t), scaling(2^-8) + Taylor(10) +
  //      8 squarings; all 16x16 products via WMMA through LDS ----
  if (tid < 32) {
    const float tv = tt[ti];
    const float w0 = omega[0];
    const float sc = tv * 0.00390625f;  // t / 2^8
#pragma unroll
    for (int e = 0; e < 8; ++e) {
      const int idx = lane * 8 + e;
      sB[idx] = (M[idx] * w0 + M0[idx]) * sc;
    }
    const int NT = 10;
#pragma unroll
    for (int e = 0; e < 8; ++e) {
      const int idx = lane * 8 + e;
      const float id = ((idx >> 4) == (idx & 15)) ? 1.0f : 0.0f;
      sT[idx] = id + sB[idx] * (1.0f / (float)NT);
    }
    for (int k = NT - 1; k >= 1; --k) {  // Horner: T = I + (B/k)*T
      v8f c = wave_mm16(sB, sT, r, h);
      frag_to_lds(sU, c, r, h);
      const float inv = 1.0f / (float)k;
#pragma unroll
      for (int e = 0; e < 8; ++e) {
        const int idx = lane * 8 + e;
        const float id = ((idx >> 4) == (idx & 15)) ? 1.0f : 0.0f;
        sT[idx] = id + sU[idx] * inv;
      }
    }
    float* Tm = sT;
    float* Um = sU;
#pragma unroll
    for (int sq = 0; sq < 8; ++sq) {  // T = T*T, 8x; even swaps -> ends in sT
      v8f c = wave_mm16(Tm, Tm, r, h);
      frag_to_lds(Um, c, r, h);
      float* tmp = Tm; Tm = Um; Um = tmp;
    }
  }
  __syncthreads();

  // ---- preload E as A-fragments (held in VGPRs for the whole tile loop) ----
  v2f afr[4];
#pragma unroll
  for (int kk = 0; kk < 4; ++kk) {
    const int kb = 4 * kk + 2 * h;
    afr[kk].x = sT[r * 16 + kb];
    afr[kk].y = sT[r * 16 + kb + 1];
  }

  // ---- streaming GEMM: out[t, n0:n0+16, :] = (E x X^T)^T ----
  const int tile0 = blockIdx.x * TPB + wave * TPW;
  for (int i = 0; i < TPW; ++i) {
    const int tile = tile0 + i;  // wave-uniform -> EXEC stays all-ones
    if (tile >= n_tiles) break;
    const int n0 = tile * 16;
    const v2f* xr = (const v2f*)(x + (size_t)(n0 + r) * 16);
    v8f c = {};
#pragma unroll
    for (int kk = 0; kk < 4; ++kk) {
      const v2f b = xr[2 * kk + h];  // B frag: x[n0+r][4kk+2h .. +1]
      c = __builtin_amdgcn_wmma_f32_16x16x4_f32(false, afr[kk], false, b,
                                                (short)0, c, false, false);
    }
    // lane writes out[t][n0+r][8h .. 8h+7]: two aligned b128 NT stores
    float* op = out + ((size_t)ti * n_points + n0 + r) * 16 + 8 * h;
    v4f lo = {c[0], c[1], c[2], c[3]};
    v4f hi = {c[4], c[5], c[6], c[7]};
    __builtin_nontemporal_store(lo, (v4f*)op);
    __builtin_nontemporal_store(hi, (v4f*)(op + 4));
  }
}

extern "C" void kernel_launch(void* const* d_in, const int* in_sizes, int n_in,
                              void* d_out, int out_size, void* d_ws,
                              size_t ws_size, hipStream_t stream) {
  (void)n_in; (void)out_size; (void)d_ws; (void)ws_size;
  const float* x  = (const float*)d_in[0];
  const float* t  = (const float*)d_in[1];
  const float* om = (const float*)d_in[2];
  const float* M  = (const float*)d_in[3];
  const float* M0 = (const float*)d_in[4];
  float* out = (float*)d_out;

  const int n_points = in_sizes[0] / 16;  // 16384
  const int n_times  = in_sizes[1];       // 512
  const int n_tiles  = (n_points + 15) / 16;

  dim3 grid((n_tiles + TPB - 1) / TPB, n_times);
  fourierflow_kernel<<<grid, dim3(256), 0, stream>>>(x, t, om, M, M0, out,
                                                     n_points, n_tiles);
}